// KVCache_7584912245141
// MI455X (gfx1250) — compile-verified
//
#include <hip/hip_runtime.h>

// Problem constants (match reference)
#define BB    4
#define HH    32
#define SS    4096
#define DD    128      // floats per row = 512 bytes
#define HCC   1024
#define SNEW  512
#define NROW  (BB*HH*HCC)   // 131072 gathered rows per K or V

typedef float v4f __attribute__((ext_vector_type(4)));

// ---- pass 1: pos_map[s] = -1 ----------------------------------------------
__global__ void pos_map_init(int* __restrict__ pos_map) {
    int i = blockIdx.x * blockDim.x + threadIdx.x;
    if (i < SS) pos_map[i] = -1;
}

// ---- pass 2: pos_map[input_pos[j]] = j  (virtual scatter) -----------------
// The harness forbids mutating d_in, so the cache update is virtualized:
// gathered rows whose seq-position was overwritten are redirected to k_val/v_val.
__global__ void pos_map_scatter(const int* __restrict__ input_pos,
                                int* __restrict__ pos_map) {
    int j = blockIdx.x * blockDim.x + threadIdx.x;
    if (j < SNEW) {
        int s = input_pos[j];
        if (s >= 0 && s < SS) pos_map[s] = j;
    }
}

// ---- pass 3: gather. One wave32 per index -> K row (512B) + V row (512B) --
// Row decode (index load, pos_map lookup, branch) is wave-uniform and done
// once per 1 KiB moved; each lane does two independent global_load_b128 and
// two non-temporal global_store_b128.
__global__ __launch_bounds__(256) void kv_gather(
    const float* __restrict__ k_cache, const float* __restrict__ v_cache,
    const float* __restrict__ k_val,   const float* __restrict__ v_val,
    const int*   __restrict__ flatten_index,
    const int*   __restrict__ pos_map,
    float*       __restrict__ out)
{
    const int lane = threadIdx.x & 31;
    // global wave id == gather index in [0, NROW)
    int idx = blockIdx.x * (blockDim.x >> 5) + (threadIdx.x >> 5);
    // wave-uniform: pin to SGPRs so all row logic goes scalar
    idx = __builtin_amdgcn_readfirstlane(idx);

    int r  = __builtin_amdgcn_readfirstlane(flatten_index[idx]); // row in (B*H*S)
    int s  = r % SS;                                             // seq position
    int bh = r / SS;                                             // fused (b*H+h)
    int j  = __builtin_amdgcn_readfirstlane(pos_map[s]);         // -1 or chunk row

    const float* __restrict__ ksrc;
    const float* __restrict__ vsrc;
    if (j >= 0) {
        // row overwritten by the new chunk -> read from k_val/v_val
        size_t o = (size_t)(bh * SNEW + j) * DD;
        ksrc = k_val + o;
        vsrc = v_val + o;
    } else {
        size_t o = (size_t)r * DD;
        ksrc = k_cache + o;
        vsrc = v_cache + o;
    }

    // two independent 128-bit loads per lane (32 lanes x 16B = one 512B row each)
    v4f kd = ((const v4f*)ksrc)[lane];
    v4f vd = ((const v4f*)vsrc)[lane];

    // streaming 128 MiB output: non-temporal stores keep the 192MB L2 free
    // for gathered cache rows / repeated indices
    __builtin_nontemporal_store(kd, (v4f*)(out + (size_t)idx * DD) + lane);
    __builtin_nontemporal_store(vd, (v4f*)(out + (size_t)(idx + NROW) * DD) + lane);
}

extern "C" void kernel_launch(void* const* d_in, const int* in_sizes, int n_in,
                              void* d_out, int out_size, void* d_ws, size_t ws_size,
                              hipStream_t stream) {
    const float* k_cache       = (const float*)d_in[0];
    const float* v_cache       = (const float*)d_in[1];
    const float* k_val         = (const float*)d_in[2];
    const float* v_val         = (const float*)d_in[3];
    const int*   input_pos     = (const int*)d_in[4];
    const int*   flatten_index = (const int*)d_in[5];
    float*       out           = (float*)d_out;

    int* pos_map = (int*)d_ws;   // SS * sizeof(int) = 16 KB of workspace

    pos_map_init<<<(SS + 255) / 256, 256, 0, stream>>>(pos_map);
    pos_map_scatter<<<(SNEW + 255) / 256, 256, 0, stream>>>(input_pos, pos_map);

    // NROW waves, 8 waves (256 threads) per block; each wave emits K+V rows
    const int waves_per_block = 256 / 32;                  // 8
    const int blocks          = NROW / waves_per_block;    // 16384
    kv_gather<<<blocks, 256, 0, stream>>>(k_cache, v_cache, k_val, v_val,
                                          flatten_index, pos_map, out);
}